// CGIterator_47648367182203
// MI455X (gfx1250) — compile-verified
//
#include <hip/hip_runtime.h>

// Fused CG-iterator for MI455X (gfx1250, wave32, WMMA).
// One workgroup = 8 points (n); everything between HBM read and HBM write
// lives in LDS (270 KB of the 320 KB WGP LDS). All K-contraction work
// (mixer linears, iteration linears, and the CG tensor product recast as a
// 16x256 GEMM) runs on v_wmma_f32_16x16x4_f32.
// A small pre-pass pair-packs the 1.25 MiB of weights into d_ws so each
// WMMA B-fragment is a single global_load_b64 instead of two b32 loads.

typedef float f32x2 __attribute__((ext_vector_type(2)));
typedef float f32x8 __attribute__((ext_vector_type(8)));

constexpr int NPTS   = 16384;
constexpr int KDIM   = 128;
constexpr int TILE_N = 8;      // points per workgroup
constexpr int RSTR   = 132;    // LDS row stride in floats (128 + 4 pad -> bank spread)
constexpr int BUF_ROWS = 160;  // 16-aligned l-sections: 16 + 32 + 48 + 64
constexpr int MC_STR = 260;    // CG matrix row stride (256 + 4 pad)
constexpr int NWAVES = 16;     // 512 threads
constexpr int WMAT   = KDIM * KDIM;     // 16384 floats per weight matrix
constexpr int NWMATS = 20;              // 12 mixer (3x4) + 8 iter (2x4)

constexpr int DEG_[4]      = {1, 3, 5, 7};
constexpr int SEC_OFF_[4]  = {0, 16, 48, 96};   // section start row (16-aligned)
constexpr int SEC_REAL_[4] = {8, 24, 40, 56};   // real rows = TILE_N * DEG[l]
// flattened m-row (0..15)  -> (l, m)
constexpr int G2L_[16] = {0, 1,1,1, 2,2,2,2,2, 3,3,3,3,3,3,3};
constexpr int G2C_[16] = {0, 0,1,2, 0,1,2,3,4, 0,1,2,3,4,5,6};
// row-tile (of 16 rows) -> which l section it belongs to
constexpr int RT_L_[10] = {0, 1,1, 2,2,2, 3,3,3,3};
// output offsets of each l block in d_out (floats)
constexpr size_t OUT_OFF_[4] = {0, (size_t)NPTS*KDIM, (size_t)NPTS*KDIM*4, (size_t)NPTS*KDIM*9};

// D = A(16x4) * B(4x16) + C(16x16), fp32 WMMA.
// A layout (ISA 7.12.2): lane&15 = M, vgpr j / lane-half = K  (K = 2*(lane>>4)+j)
// B layout (assumed mirror): lane&15 = N, K = 2*(lane>>4)+j
// C layout: vgpr r -> M = r + 8*(lane>>4), N = lane&15
__device__ __forceinline__ f32x8 wmma4(f32x2 a, f32x2 b, f32x8 c) {
  return __builtin_amdgcn_wmma_f32_16x16x4_f32(false, a, false, b, (short)0, c, false, false);
}

// Weight pre-pack: WT[(k>>1)*256 + q*2 + (k&1)] = W[k][q].
// The two K-adjacent values a lane needs for a B fragment become one b64 load.
__global__ __launch_bounds__(256)
void pack_weights(const float* __restrict__ mixer_w, const float* __restrict__ iter_w,
                  float* __restrict__ wt) {
  const int e = blockIdx.x * 256 + threadIdx.x;          // over NWMATS*WMAT elements
  const int m = e >> 14, within = e & (WMAT - 1);
  const int k = within >> 7, q = within & 127;
  const float v = (m < 12) ? mixer_w[(size_t)m * WMAT + within]
                           : iter_w[(size_t)(m - 12) * WMAT + within];
  wt[(size_t)m * WMAT + (k >> 1) * 256 + q * 2 + (k & 1)] = v;
}

// Column-chunk [ct0,ct1) of one 16-row output block of
// OUT = A_lds(160x128) @ W(128x128) [+= if accum], weights pair-packed.
// The 16x128 A strip is held in registers, so OUT may alias A
// (used for the in-place density -> mixed2 pass).
__device__ __forceinline__ void gemm_rowtile(const float* __restrict__ A,
                                             float* __restrict__ O,
                                             const float* __restrict__ WT,
                                             int rt, int ct0, int ct1,
                                             bool accum, int lane) {
  const int lo = lane & 15, hi = lane >> 4;
  f32x2 a[32];
#pragma unroll
  for (int kc = 0; kc < 32; ++kc)   // K chunks of 4
    a[kc] = *(const f32x2*)&A[(rt * 16 + lo) * RSTR + kc * 4 + 2 * hi];

  const int l = RT_L_[rt], soff = SEC_OFF_[l], sreal = SEC_REAL_[l];
  for (int ct = ct0; ct < ct1; ++ct) {  // column tiles of 16
    f32x8 acc = {0.f, 0.f, 0.f, 0.f, 0.f, 0.f, 0.f, 0.f};
    if (accum) {
#pragma unroll
      for (int r = 0; r < 8; ++r)
        acc[r] = O[(rt * 16 + r + 8 * hi) * RSTR + ct * 16 + lo];
    }
#pragma unroll
    for (int kc = 0; kc < 32; ++kc) {
      // pair-packed: rows (kc*4+2*hi, +1) at consecutive floats -> one b64
      const f32x2 b = *(const f32x2*)&WT[(kc * 2 + hi) * 256 + (ct * 16 + lo) * 2];
      acc = wmma4(a[kc], b, acc);
    }
#pragma unroll
    for (int r = 0; r < 8; ++r) {
      const int row = rt * 16 + r + 8 * hi;
      if (row - soff < sreal)                     // drop padding rows
        O[row * RSTR + ct * 16 + lo] = acc[r];
    }
  }
}

// CG tensor product for one (n_local, 16-column) task, as tp = Mcg(16x256) @ prod.
// prod[p] = cur[p>>4] * mix[p&15], built lane-wise into the B fragment.
// Result overwrites the mix buffer region (column-exclusive -> race-free).
__device__ __forceinline__ void tp_task(const float* __restrict__ CUR,
                                        float* __restrict__ MIX,
                                        const float* __restrict__ MCG,
                                        int n_local, int k0, int lane) {
  const int lo = lane & 15, hi = lane >> 4;
  const int col = k0 + lo;
  float av[16], bv[16];
  int rowg[16];
#pragma unroll
  for (int g = 0; g < 16; ++g) {
    const int row = SEC_OFF_[G2L_[g]] + n_local * DEG_[G2L_[g]] + G2C_[g];
    rowg[g] = row;
    av[g] = CUR[row * RSTR + col];
    bv[g] = MIX[row * RSTR + col];
  }
  f32x8 acc = {0.f, 0.f, 0.f, 0.f, 0.f, 0.f, 0.f, 0.f};
#pragma unroll
  for (int q = 0; q < 64; ++q) {                 // 256 products / 4 per WMMA
    const f32x2 af = *(const f32x2*)&MCG[lo * MC_STR + 4 * q + 2 * hi];
    f32x2 bf;
#pragma unroll
    for (int j = 0; j < 2; ++j) {
      const int pl = 4 * q + j, ph = 4 * q + 2 + j;   // low/high lane-half rows
      const float vl = av[pl >> 4] * bv[pl & 15];
      const float vh = av[ph >> 4] * bv[ph & 15];
      bf[j] = hi ? vh : vl;
    }
    acc = wmma4(af, bf, acc);
  }
#pragma unroll
  for (int r = 0; r < 8; ++r) {
    const int row = hi ? rowg[r + 8] : rowg[r];
    MIX[row * RSTR + col] = acc[r];
  }
}

__global__ __launch_bounds__(512, 1)
void cg_iter_fused(const float* __restrict__ f0, const float* __restrict__ f1,
                   const float* __restrict__ f2, const float* __restrict__ f3,
                   const float* __restrict__ wt, const float* __restrict__ cg,
                   float* __restrict__ out) {
  __shared__ float buf[3][BUF_ROWS * RSTR];   // density->mixed2 / current / mixedB
  __shared__ float mcg[16 * MC_STR];          // dense 16x256 CG matrix

  const int tid = threadIdx.x, lane = tid & 31, wave = tid >> 5;
  const int n_base = blockIdx.x * TILE_N;
  const float* fin[4] = {f0, f1, f2, f3};

  // ---- stage 0: load density tile into buf0, gather CG matrix ----
#pragma unroll
  for (int l = 0; l < 4; ++l) {
    const float* src = fin[l] + (size_t)n_base * DEG_[l] * KDIM;
    const int nf4 = SEC_REAL_[l] * (KDIM / 4);
    for (int v = tid; v < nf4; v += 512) {
      const int row = v >> 5, c = (v & 31) * 4;
      *(float4*)&buf[0][(SEC_OFF_[l] + row) * RSTR + c] = *(const float4*)(src + row * KDIM + c);
    }
  }
  // zero the 8 padding rows per section (they feed WMMA A fragments)
  for (int v = tid; v < 4 * 8 * RSTR; v += 512) {
    const int l = v / (8 * RSTR), rem = v - l * 8 * RSTR;
    buf[0][(SEC_OFF_[l] + SEC_REAL_[l] + rem / RSTR) * RSTR + rem % RSTR] = 0.f;
  }
  // Mcg[o][p] = cg[l1,l2,l,i,j,m]; invalid (l1,l2,l) triples are exact zeros
  // in the input (mask applied host-side), so a dense gather is correct.
  for (int e = tid; e < 16 * 256; e += 512) {
    const int o = e >> 8, p = e & 255, ig = p >> 4, jg = p & 15;
    const int l1 = G2L_[ig], i = G2C_[ig], l2 = G2L_[jg], j = G2C_[jg];
    const int l = G2L_[o], m = G2C_[o];
    mcg[o * MC_STR + p] = cg[((((l1 * 4 + l2) * 4 + l) * 7 + i) * 7 + j) * 7 + m];
  }
  __syncthreads();

  // ---- stage 1: buf1 = D @ mixer_w[0] (current), buf2 = D @ mixer_w[1] ----
  // tasks = 2 gemms x 10 row-tiles x 2 column-halves = 40
  for (int tt = wave; tt < 40; tt += NWAVES) {
    const int g = tt / 20, rem = tt % 20, rt = rem >> 1, ch = rem & 1;
    gemm_rowtile(buf[0], buf[1 + g], wt + (size_t)(g * 4 + RT_L_[rt]) * WMAT,
                 rt, ch * 4, ch * 4 + 4, false, lane);
  }
  __syncthreads();
  // ---- stage 2: buf0 = D @ mixer_w[2] in place (row-blocks are row-local) ----
  for (int tt = wave; tt < 20; tt += NWAVES) {
    const int rt = tt >> 1, ch = tt & 1;
    gemm_rowtile(buf[0], buf[0], wt + (size_t)(8 + RT_L_[rt]) * WMAT,
                 rt, ch * 4, ch * 4 + 4, false, lane);
  }
  __syncthreads();

  // ---- iterations: tp = TP(current, mixed[t+1]); current += tp @ iter_w[t] ----
  for (int t = 0; t < 2; ++t) {
    float* mixb = (t == 0) ? buf[2] : buf[0];
    for (int tt = wave; tt < 64; tt += NWAVES)          // 8 n x 8 col-groups
      tp_task(buf[1], mixb, mcg, tt >> 3, (tt & 7) * 16, lane);
    __syncthreads();
    for (int tt = wave; tt < 20; tt += NWAVES) {
      const int rt = tt >> 1, ch = tt & 1;
      gemm_rowtile(mixb, buf[1], wt + (size_t)(12 + t * 4 + RT_L_[rt]) * WMAT,
                   rt, ch * 4, ch * 4 + 4, true, lane);
    }
    __syncthreads();
  }

  // ---- store current (buf1) to d_out ----
#pragma unroll
  for (int l = 0; l < 4; ++l) {
    float* dst = out + OUT_OFF_[l] + (size_t)n_base * DEG_[l] * KDIM;
    const int nf4 = SEC_REAL_[l] * (KDIM / 4);
    for (int v = tid; v < nf4; v += 512) {
      const int row = v >> 5, c = (v & 31) * 4;
      *(float4*)(dst + row * KDIM + c) = *(const float4*)&buf[1][(SEC_OFF_[l] + row) * RSTR + c];
    }
  }
}

extern "C" void kernel_launch(void* const* d_in, const int* in_sizes, int n_in,
                              void* d_out, int out_size, void* d_ws, size_t ws_size,
                              hipStream_t stream) {
  (void)in_sizes; (void)n_in; (void)out_size; (void)ws_size;
  const float* f0 = (const float*)d_in[0];
  const float* f1 = (const float*)d_in[1];
  const float* f2 = (const float*)d_in[2];
  const float* f3 = (const float*)d_in[3];
  const float* mixer_w = (const float*)d_in[4];
  const float* iter_w  = (const float*)d_in[5];
  const float* cg      = (const float*)d_in[6];
  float* wt = (float*)d_ws;   // NWMATS * WMAT * 4 = 1.25 MiB of scratch

  pack_weights<<<(NWMATS * WMAT) / 256, 256, 0, stream>>>(mixer_w, iter_w, wt);
  cg_iter_fused<<<NPTS / TILE_N, 512, 0, stream>>>(f0, f1, f2, f3, wt, cg, (float*)d_out);
}